// ModelNew_23983097380969
// MI455X (gfx1250) — compile-verified
//
#include <hip/hip_runtime.h>

// Reverse cumulative sum (suffix sum) along axis 1 of a 4096x32768 fp32 matrix.
// Bandwidth-bound: 1 GB traffic / 23.3 TB/s ~= 45 us floor.
// One workgroup (1024 threads = 32 wave32 waves) per row; row staged in LDS via
// CDNA5 async global<->LDS copies (ASYNCcnt), scan done per-thread in registers.

#define ROW_N      32768
#define BLOCK_T    1024
#define EPT        32          // elements per thread
#define F4PT       8           // float4s per thread
#define PADF4      9           // padded float4 slots per thread (36 floats, 16B aligned)
#define NWAVES     (BLOCK_T / 32)

__global__ __launch_bounds__(BLOCK_T)
void suffix_scan_rows_kernel(const float* __restrict__ x,
                             float* __restrict__ out) {
    // Row data, padded: thread t owns float4 slots [t*9, t*9+8); slot t*9+8 is pad.
    __shared__ float4 s4[BLOCK_T * PADF4];   // 147,456 bytes (row fits in 320KB WGP LDS)
    __shared__ float  wtot[NWAVES];          // per-wave totals

    const unsigned tid  = threadIdx.x;
    const unsigned lane = tid & 31u;
    const unsigned wave = tid >> 5;

    const float4* __restrict__ src4 =
        (const float4*)(x + (size_t)blockIdx.x * ROW_N);
    float4* __restrict__ dst4 =
        (float4*)(out + (size_t)blockIdx.x * ROW_N);

    // 32-bit LDS byte offset of s4 (low 32 bits of the generic pointer).
    const unsigned ldsBase = (unsigned)(size_t)(&s4[0]);

    // ---- Phase 1: async copy row -> LDS (global side coalesced, LDS side padded) ----
    // global float4 index g4 -> padded float4 slot (g4/8)*9 + g4%8 == g4 + (g4>>3)
#pragma unroll
    for (int it = 0; it < F4PT; ++it) {
        unsigned g4      = (unsigned)it * BLOCK_T + tid;
        unsigned ldsAddr = ldsBase + ((g4 + (g4 >> 3)) << 4);
        const float4* gaddr = src4 + g4;
        asm volatile("global_load_async_to_lds_b128 %0, %1, off"
                     :: "v"(ldsAddr), "v"(gaddr)
                     : "memory");
    }
    asm volatile("s_wait_asynccnt 0x0" ::: "memory");
    __syncthreads();

    // ---- Phase 2: per-thread suffix scan of own contiguous 32-element chunk ----
    float v[EPT];
    float4* myChunk = &s4[tid * PADF4];
#pragma unroll
    for (int j = 0; j < F4PT; ++j) {
        float4 q = myChunk[j];
        v[4 * j + 0] = q.x;
        v[4 * j + 1] = q.y;
        v[4 * j + 2] = q.z;
        v[4 * j + 3] = q.w;
    }
#pragma unroll
    for (int j = EPT - 2; j >= 0; --j) v[j] += v[j + 1];
    const float total = v[0];

    // ---- Phase 3: block-wide exclusive suffix scan of per-thread totals ----
    // Wave-level inclusive suffix scan via lane shuffles (wave32).
    float sfx = total;
#pragma unroll
    for (int d = 1; d < 32; d <<= 1) {
        float o = __shfl_down(sfx, (unsigned)d, 32);
        if (lane + (unsigned)d < 32u) sfx += o;
    }
    float excl = __shfl_down(sfx, 1u, 32);   // suffix of lanes strictly after me
    if (lane == 31u) excl = 0.0f;
    if (lane == 0u) wtot[wave] = sfx;        // whole-wave total
    __syncthreads();

    float off = excl;
    for (unsigned w2 = wave + 1u; w2 < NWAVES; ++w2) off += wtot[w2];

    // ---- Phase 4: add offset, write results back into own padded LDS chunk ----
#pragma unroll
    for (int j = 0; j < F4PT; ++j) {
        float4 q;
        q.x = v[4 * j + 0] + off;
        q.y = v[4 * j + 1] + off;
        q.z = v[4 * j + 2] + off;
        q.w = v[4 * j + 3] + off;
        myChunk[j] = q;
    }
    __syncthreads();

    // ---- Phase 5: async store LDS -> global (global side coalesced) ----
#pragma unroll
    for (int it = 0; it < F4PT; ++it) {
        unsigned g4      = (unsigned)it * BLOCK_T + tid;
        unsigned ldsAddr = ldsBase + ((g4 + (g4 >> 3)) << 4);
        float4* gaddr = dst4 + g4;
        asm volatile("global_store_async_from_lds_b128 %0, %1, off"
                     :: "v"(gaddr), "v"(ldsAddr)
                     : "memory");
    }
    asm volatile("s_wait_asynccnt 0x0" ::: "memory");
}

extern "C" void kernel_launch(void* const* d_in, const int* in_sizes, int n_in,
                              void* d_out, int out_size, void* d_ws, size_t ws_size,
                              hipStream_t stream) {
    (void)n_in; (void)d_ws; (void)ws_size; (void)out_size;
    const float* x = (const float*)d_in[0];
    float* out = (float*)d_out;
    const int rows = in_sizes[0] / ROW_N;    // 4096
    suffix_scan_rows_kernel<<<dim3(rows), dim3(BLOCK_T), 0, stream>>>(x, out);
}